// NAM_19112604467534
// MI455X (gfx1250) — compile-verified
//
#include <hip/hip_runtime.h>
#include <math.h>

#define B_SZ  2048
#define F_SZ  256
#define U_SZ  64
#define H1_SZ 64
#define H2_SZ 32
#define E_SZ  257

typedef float v2f __attribute__((ext_vector_type(2)));
typedef float v8f __attribute__((ext_vector_type(8)));

__device__ __forceinline__ float relu_f(float v) { return fmaxf(v, 0.0f); }

// -------------------------------------------------------------------------
// Kernel 1: batch-independent attention precompute.
//   blocks 0..255 : M[i][j] = (Wq col i + bq) . (Wk col j + bk)
//   block 256     : p[j], r[j], s[j], and scalars t, m, ob
// -------------------------------------------------------------------------
__global__ __launch_bounds__(256) void attn_pre_kernel(
    const float* __restrict__ in_w, const float* __restrict__ in_b,
    const float* __restrict__ out_w, const float* __restrict__ out_b,
    float* __restrict__ M, float* __restrict__ p, float* __restrict__ r,
    float* __restrict__ sv, float* __restrict__ scal) {
  const int blk = blockIdx.x;
  const int tid = threadIdx.x;

  if (blk < F_SZ) {
    __shared__ float a_sh[E_SZ];
    for (int e = tid; e < E_SZ; e += 256)
      a_sh[e] = in_w[(size_t)e * E_SZ + blk] + in_b[e];
    __syncthreads();
    const int j = tid;
    float acc = 0.0f;
    for (int e = 0; e < E_SZ; ++e) {
      float d = in_w[(size_t)(E_SZ + e) * E_SZ + j] + in_b[E_SZ + e];
      acc = fmaf(a_sh[e], d, acc);
    }
    M[blk * F_SZ + j] = acc;
  } else {
    const int j = tid;
    float accp = 0.0f, accr = 0.0f, accs = 0.0f;
    for (int e = 0; e < E_SZ; ++e) {
      float a = in_w[(size_t)e * E_SZ + j] + in_b[e];
      float g = in_w[(size_t)(E_SZ + e) * E_SZ + (E_SZ - 1)];
      accp = fmaf(a, g, accp);
      float d = in_w[(size_t)(E_SZ + e) * E_SZ + j] + in_b[E_SZ + e];
      float c = in_w[(size_t)e * E_SZ + (E_SZ - 1)];
      accr = fmaf(c, d, accr);
      float h = in_w[(size_t)(2 * E_SZ + e) * E_SZ + j] + in_b[2 * E_SZ + e];
      float ow = out_w[(size_t)(E_SZ - 1) * E_SZ + e];
      accs = fmaf(h, ow, accs);
    }
    p[j] = accp; r[j] = accr; sv[j] = accs;
    if (j == 0) {
      float tAcc = 0.0f, mAcc = 0.0f;
      for (int e = 0; e < E_SZ; ++e) {
        float c  = in_w[(size_t)e * E_SZ + (E_SZ - 1)];
        float g  = in_w[(size_t)(E_SZ + e) * E_SZ + (E_SZ - 1)];
        float u  = in_w[(size_t)(2 * E_SZ + e) * E_SZ + (E_SZ - 1)];
        float ow = out_w[(size_t)(E_SZ - 1) * E_SZ + e];
        tAcc = fmaf(c, g, tAcc);
        mAcc = fmaf(u, ow, mAcc);
      }
      scal[0] = tAcc;
      scal[1] = mAcc;
      scal[2] = out_b[E_SZ - 1];
    }
  }
}

// -------------------------------------------------------------------------
// Kernel 2: fused per-(b,i) softmax + context projection.
// One block per batch row b; one wave per attention row i (round-robin).
// Prefetches the wave's next M row (global_prefetch_b8) while working.
// Writes xaT transposed [F, B] for contiguous per-feature access later.
// -------------------------------------------------------------------------
__global__ __launch_bounds__(256) void attn_apply_kernel(
    const float* __restrict__ x, const float* __restrict__ M,
    const float* __restrict__ p, const float* __restrict__ r,
    const float* __restrict__ sv, const float* __restrict__ scal,
    float* __restrict__ xaT) {
  const float INV_SQRT_E = 0.062378286f;  // 1/sqrt(257)
  __shared__ float xs[F_SZ], rs[F_SZ], ss[F_SZ], ps[F_SZ];
  const int b = blockIdx.x;
  const int tid = threadIdx.x;
  xs[tid] = x[(size_t)b * F_SZ + tid];
  rs[tid] = r[tid];
  ss[tid] = sv[tid];
  ps[tid] = p[tid];
  __syncthreads();
  const float tS = scal[0], mS = scal[1], obS = scal[2];
  const int wave = tid >> 5, lane = tid & 31;

  for (int i = wave; i < F_SZ; i += 8) {
    // prefetch the next M row this wave will touch (32 lanes x 32B = 1KB row)
    if (i + 8 < F_SZ)
      __builtin_prefetch(&M[(size_t)(i + 8) * F_SZ + lane * 8], 0, 0);

    const float xi = xs[i];
    const float pi = ps[i];
    const float* Mrow = M + (size_t)i * F_SZ;
    float sc[8];
    float mx = -3.0e38f;
#pragma unroll
    for (int q = 0; q < 8; ++q) {
      int j = lane + q * 32;
      float xj = xs[j];
      float s0 = (Mrow[j] + pi * xj + xi * (rs[j] + tS * xj)) * INV_SQRT_E;
      sc[q] = s0;
      mx = fmaxf(mx, s0);
    }
#pragma unroll
    for (int off = 16; off > 0; off >>= 1) mx = fmaxf(mx, __shfl_xor(mx, off, 32));
    float se = 0.0f, wsum = 0.0f;
#pragma unroll
    for (int q = 0; q < 8; ++q) {
      int j = lane + q * 32;
      float e = __expf(sc[q] - mx);
      se += e;
      wsum = fmaf(e, ss[j] + xs[j] * mS, wsum);
    }
#pragma unroll
    for (int off = 16; off > 0; off >>= 1) {
      se   += __shfl_xor(se, off, 32);
      wsum += __shfl_xor(wsum, off, 32);
    }
    if (lane == 0) xaT[(size_t)i * B_SZ + b] = wsum / se + obS;
  }
}

// -------------------------------------------------------------------------
// Kernel 3: per-feature MLPs via fp32 WMMA (V_WMMA_F32_16X16X4_F32).
// One block per feature; one wave handles a 16-row batch chunk at a time.
// All 16 per-chunk xa scalars are preloaded into registers so the WMMA
// loop body contains no global loads (removes per-chunk loadcnt stall).
//   layer1: rank-1 relu(xa*w1+b1) built directly into A fragments
//   layer2: [16,64]x[64,64] -> 4 N-tiles x 16 K-steps of WMMA
//   layer3: [16,64]x[64,32] -> 2 N-tiles x 16 K-steps of WMMA
//   layer4: dot with w4 via LDS staging
// -------------------------------------------------------------------------
__global__ __launch_bounds__(256) void feature_nns_kernel(
    const float* __restrict__ xaT,
    const float* __restrict__ w1, const float* __restrict__ b1,
    const float* __restrict__ w2, const float* __restrict__ b2,
    const float* __restrict__ w3, const float* __restrict__ b3,
    const float* __restrict__ w4, const float* __restrict__ b4,
    float* __restrict__ fo) {
  const int f = blockIdx.x;
  const int tid = threadIdx.x;

  __shared__ float w1s[U_SZ], b1s[U_SZ], b2s[H1_SZ], b3s[H2_SZ], w4s[H2_SZ];
  __shared__ float w2s[U_SZ * H1_SZ];   // 64x64
  __shared__ float w3s[H1_SZ * H2_SZ];  // 64x32
  __shared__ float stage[8][16 * 65];   // per-wave staging, stride-65 padded
  __shared__ float b4s;

  if (tid < U_SZ) {
    w1s[tid] = w1[(size_t)f * U_SZ + tid];
    b1s[tid] = b1[(size_t)f * U_SZ + tid];
    b2s[tid] = b2[(size_t)f * H1_SZ + tid];
  }
  if (tid < H2_SZ) {
    b3s[tid] = b3[(size_t)f * H2_SZ + tid];
    w4s[tid] = w4[(size_t)f * H2_SZ + tid];
  }
  if (tid == 0) b4s = b4[f];
  for (int idx = tid; idx < U_SZ * H1_SZ; idx += 256)
    w2s[idx] = w2[(size_t)f * U_SZ * H1_SZ + idx];
  for (int idx = tid; idx < H1_SZ * H2_SZ; idx += 256)
    w3s[idx] = w3[(size_t)f * H1_SZ * H2_SZ + idx];
  __syncthreads();

  const int wave = tid >> 5;
  const int lane = tid & 31;
  const int half = lane >> 4;   // k-pair select (ISA A/B fragment layout)
  const int lrow = lane & 15;   // M row (A) / N column (B, C, D)
  float* st = stage[wave];

  // Preload this wave's 16 chunk scalars (software pipelining: single wait).
  const int NCHUNK = (B_SZ / 16) / 8;  // 16
  float xv_all[NCHUNK];
#pragma unroll
  for (int c = 0; c < NCHUNK; ++c) {
    const int b0 = (wave + c * 8) * 16;
    xv_all[c] = xaT[(size_t)f * B_SZ + b0 + lrow];
  }

#pragma unroll 1
  for (int c = 0; c < NCHUNK; ++c) {
    const int b0 = (wave + c * 8) * 16;
    const float xv = xv_all[c];

    // ---- layer 2 accumulators seeded with b2 (C/D layout: col = nb*16+lrow)
    v8f acc2[4];
#pragma unroll
    for (int nb = 0; nb < 4; ++nb) {
      float bb = b2s[nb * 16 + lrow];
#pragma unroll
      for (int rr = 0; rr < 8; ++rr) acc2[nb][rr] = bb;
    }
#pragma unroll
    for (int kb = 0; kb < 16; ++kb) {
      const int k0 = kb * 4 + half * 2;
      v2f a;
      a.x = relu_f(fmaf(xv, w1s[k0],     b1s[k0]));
      a.y = relu_f(fmaf(xv, w1s[k0 + 1], b1s[k0 + 1]));
#pragma unroll
      for (int nb = 0; nb < 4; ++nb) {
        v2f bf;
        bf.x = w2s[(k0)     * H1_SZ + nb * 16 + lrow];
        bf.y = w2s[(k0 + 1) * H1_SZ + nb * 16 + lrow];
        acc2[nb] = __builtin_amdgcn_wmma_f32_16x16x4_f32(
            false, a, false, bf, (short)0, acc2[nb], false, false);
      }
    }

    // ---- relu(h2) -> LDS staging (D layout row = rr + 8*half)
#pragma unroll
    for (int nb = 0; nb < 4; ++nb)
#pragma unroll
      for (int rr = 0; rr < 8; ++rr)
        st[(rr + 8 * half) * 65 + nb * 16 + lrow] = relu_f(acc2[nb][rr]);

    // ---- layer 3
    v8f acc3[2];
#pragma unroll
    for (int nb = 0; nb < 2; ++nb) {
      float bb = b3s[nb * 16 + lrow];
#pragma unroll
      for (int rr = 0; rr < 8; ++rr) acc3[nb][rr] = bb;
    }
#pragma unroll
    for (int kb = 0; kb < 16; ++kb) {
      const int k0 = kb * 4 + half * 2;
      v2f a;
      a.x = st[lrow * 65 + k0];
      a.y = st[lrow * 65 + k0 + 1];
#pragma unroll
      for (int nb = 0; nb < 2; ++nb) {
        v2f bf;
        bf.x = w3s[(k0)     * H2_SZ + nb * 16 + lrow];
        bf.y = w3s[(k0 + 1) * H2_SZ + nb * 16 + lrow];
        acc3[nb] = __builtin_amdgcn_wmma_f32_16x16x4_f32(
            false, a, false, bf, (short)0, acc3[nb], false, false);
      }
    }

    // ---- relu(h3) -> staging, then layer 4 dot with w4
#pragma unroll
    for (int nb = 0; nb < 2; ++nb)
#pragma unroll
      for (int rr = 0; rr < 8; ++rr)
        st[(rr + 8 * half) * 65 + nb * 16 + lrow] = relu_f(acc3[nb][rr]);

    float partial = 0.0f;
#pragma unroll
    for (int g = 0; g < 16; ++g) {
      const int gg = half * 16 + g;
      partial = fmaf(st[lrow * 65 + gg], w4s[gg], partial);
    }
    partial += __shfl_xor(partial, 16, 32);
    if (half == 0)
      fo[(size_t)(b0 + lrow) * F_SZ + f] = partial + b4s;
  }
}

// -------------------------------------------------------------------------
// Kernel 4: out[b] = sum_f fo[b,f] + bias
// -------------------------------------------------------------------------
__global__ __launch_bounds__(256) void reduce_out_kernel(
    const float* __restrict__ fo, const float* __restrict__ bias,
    float* __restrict__ out) {
  const int b = blockIdx.x;
  const int tid = threadIdx.x;
  float v = fo[(size_t)b * F_SZ + tid];
#pragma unroll
  for (int off = 16; off > 0; off >>= 1) v += __shfl_xor(v, off, 32);
  __shared__ float wsum[8];
  if ((tid & 31) == 0) wsum[tid >> 5] = v;
  __syncthreads();
  if (tid < 8) {
    float t = wsum[tid];
#pragma unroll
    for (int off = 4; off > 0; off >>= 1) t += __shfl_xor(t, off, 32);
    if (tid == 0) out[b] = t + bias[0];
  }
}

// -------------------------------------------------------------------------
extern "C" void kernel_launch(void* const* d_in, const int* in_sizes, int n_in,
                              void* d_out, int out_size, void* d_ws, size_t ws_size,
                              hipStream_t stream) {
  const float* x     = (const float*)d_in[0];
  const float* w1    = (const float*)d_in[1];
  const float* b1    = (const float*)d_in[2];
  const float* w2    = (const float*)d_in[3];
  const float* b2    = (const float*)d_in[4];
  const float* w3    = (const float*)d_in[5];
  const float* b3    = (const float*)d_in[6];
  const float* w4    = (const float*)d_in[7];
  const float* b4    = (const float*)d_in[8];
  const float* in_w  = (const float*)d_in[9];
  const float* in_b  = (const float*)d_in[10];
  const float* out_w = (const float*)d_in[11];
  const float* out_b = (const float*)d_in[12];
  const float* bias  = (const float*)d_in[13];

  float* out_vec = (float*)d_out;            // [B]
  float* fo      = (float*)d_out + B_SZ;     // [B, F]

  float* ws   = (float*)d_ws;
  float* xaT  = ws;                          // [F, B]
  float* M    = xaT + (size_t)F_SZ * B_SZ;   // [F, F]
  float* p    = M + F_SZ * F_SZ;             // [F]
  float* r    = p + F_SZ;                    // [F]
  float* sv   = r + F_SZ;                    // [F]
  float* scal = sv + F_SZ;                   // t, m, ob

  attn_pre_kernel<<<F_SZ + 1, 256, 0, stream>>>(in_w, in_b, out_w, out_b, M, p, r, sv, scal);
  attn_apply_kernel<<<B_SZ, 256, 0, stream>>>(x, M, p, r, sv, scal, xaT);
  feature_nns_kernel<<<F_SZ, 256, 0, stream>>>(xaT, w1, b1, w2, b2, w3, b3, w4, b4, fo);
  reduce_out_kernel<<<B_SZ, 256, 0, stream>>>(fo, bias, out_vec);
}